// DescrptSeA_403726926074
// MI455X (gfx1250) — compile-verified
//
#include <hip/hip_runtime.h>

// ---------------------------------------------------------------------------
// DescrptSeA fused kernel for gfx1250 (MI455X), fp32 WMMA path, round 2.
// One 128-thread workgroup (4 wave32) per local atom.
// All WMMA operands come from zero-padded LDS buffers via contiguous
// ds_load_b64 (no conditional global loads inside K-loops).
// ---------------------------------------------------------------------------

typedef __attribute__((ext_vector_type(2))) float v2f;
typedef __attribute__((ext_vector_type(8))) float v8f;

namespace {
constexpr int kNF   = 2;
constexpr int kNLOC = 4096;
constexpr int kNALL = 8192;
constexpr int kNNEI = 138;     // SEL = [46, 92]
constexpr int kS0   = 46;
constexpr int kS1   = 92;
constexpr int kK2   = 25;      // layer2 K (=NEURON[0])
constexpr int kN2   = 50;      // layer2 N (=NEURON[1])
constexpr int kK3   = 50;      // layer3 K
constexpr int kM    = 100;     // layer3 N (=NEURON[2])
constexpr int kKP2  = 32;      // padded K2
constexpr int kNP2  = 64;      // padded N2
constexpr int kKP3  = 64;      // padded K3
constexpr int kMP   = 112;     // padded M (7 tiles of 16)
constexpr int kSP   = 96;      // max padded neighbor rows (92 -> 96)
constexpr int kAXIS = 16;
constexpr float kRCUT = 6.0f;
constexpr float kRSM  = 0.5f;
constexpr float kPROT = 1e-6f;
}

__device__ __forceinline__ v8f wmma_f32(v2f a, v2f b, v8f c) {
  // D = A(16x4, f32) * B(4x16, f32) + C(16x16, f32)
  return __builtin_amdgcn_wmma_f32_16x16x4_f32(
      /*neg_a=*/false, a, /*neg_b=*/false, b,
      /*c_mod=*/(short)0, c, /*reuse_a=*/false, /*reuse_b=*/false);
}

__global__ __launch_bounds__(128)
void descrpt_sea_fused(const int*   __restrict__ nlist,
                       const float* __restrict__ coord,
                       const int*   __restrict__ atype,
                       const float* __restrict__ mean,
                       const float* __restrict__ stdv,
                       const float* __restrict__ w0, const float* __restrict__ b0,
                       const float* __restrict__ w1, const float* __restrict__ b1,
                       const float* __restrict__ w2, const float* __restrict__ b2,
                       float* __restrict__ out)
{
  __shared__ float DM[kNNEI * 4];       //  2.2 KB  env matrix (normalized)
  __shared__ float X1[kSP][kKP2];       // 12.0 KB  layer-1 activations
  __shared__ float X2[kSP][kKP3];       // 24.0 KB  layer-2 activations
  __shared__ float GTT[16][kSP];        //  6.0 KB  gg tile, transposed (col-major)
  __shared__ float XT[kMP][16];         //  7.0 KB  xyz^T accumulator (100x4 live)
  __shared__ float WT1[kNP2][kKP2];     //  8.0 KB  W1^T, zero-padded
  __shared__ float WT2[kMP][kKP3];      // 28.0 KB  W2^T, zero-padded
  __shared__ float RT[16][kSP];         //  6.0 KB  rr^T, rows 4..15 zero
  __shared__ float BB1[kNP2];           //  bias1, zero-padded
  __shared__ float BB2[kMP];            //  bias2, zero-padded

  const int g    = blockIdx.x;               // atom id in [0, NF*NLOC)
  const int f    = g / kNLOC;
  const int i    = g - f * kNLOC;
  const int tid  = threadIdx.x;
  const int lane = tid & 31;
  const int wave = tid >> 5;
  const int hi   = (lane >= 16) ? 1 : 0;     // half-wave selector (K rows 2,3)
  const int l16  = lane & 15;

  const float cx = coord[(size_t)(f * kNALL + i) * 3 + 0];
  const float cy = coord[(size_t)(f * kNALL + i) * 3 + 1];
  const float cz = coord[(size_t)(f * kNALL + i) * 3 + 2];
  const int   tc = atype[f * kNALL + i];     // central atom type (0/1)

  // ------------------------------------------------------------------ env mat
  for (int j = tid; j < kNNEI; j += 128) {
    const int  nl  = nlist[(size_t)g * kNNEI + j];
    const bool msk = nl >= 0;
    const int  idx = msk ? nl : 0;
    const float dx = coord[(size_t)(f * kNALL + idx) * 3 + 0] - cx;
    const float dy = coord[(size_t)(f * kNALL + idx) * 3 + 1] - cy;
    const float dz = coord[(size_t)(f * kNALL + idx) * 3 + 2] - cz;
    float d = sqrtf(dx * dx + dy * dy + dz * dz);
    d = msk ? d : 1.0f;
    const float uu = (d - kRSM) / (kRCUT - kRSM);
    const float vv = uu * uu * uu * (-6.0f * uu * uu + 15.0f * uu - 10.0f) + 1.0f;
    float w = (d <= kRSM) ? 1.0f : ((d >= kRCUT) ? 0.0f : vv);
    w = msk ? w : 0.0f;
    const float inv  = 1.0f / (d + kPROT);
    const float inv2 = inv * inv;
    const float e0 = inv * w;
    const float e1 = dx * inv2 * w;
    const float e2 = dy * inv2 * w;
    const float e3 = dz * inv2 * w;
    const float* mr = mean + ((size_t)tc * kNNEI + j) * 4;
    const float* sr = stdv + ((size_t)tc * kNNEI + j) * 4;
    DM[j * 4 + 0] = (e0 - mr[0]) / sr[0];
    DM[j * 4 + 1] = (e1 - mr[1]) / sr[1];
    DM[j * 4 + 2] = (e2 - mr[2]) / sr[2];
    DM[j * 4 + 3] = (e3 - mr[3]) / sr[3];
  }
  for (int e = tid; e < kMP * 16; e += 128) (&XT[0][0])[e] = 0.0f;
  __syncthreads();

  // ------------------------------------------------------------- per-type MLP
  for (int t = 0; t < 2; ++t) {
    const int sec  = (t == 0) ? 0 : kS0;
    const int st   = (t == 0) ? kS0 : kS1;      // live neighbor rows
    const int spad = (t == 0) ? 48 : kSP;       // padded rows
    const int mt   = spad >> 4;                 // row tiles (3 or 6)
    const float* W0t = w0 + (size_t)t * kK2;          // (1,25)
    const float* B0t = b0 + (size_t)t * kK2;
    const float* W1t = w1 + (size_t)t * kK2 * kN2;    // (25,50)
    const float* B1t = b1 + (size_t)t * kN2;
    const float* W2t = w2 + (size_t)t * kK3 * kM;     // (50,100)
    const float* B2t = b2 + (size_t)t * kM;

    // ---- stage weights (transposed, zero-padded) + biases + rr^T into LDS
    for (int e = tid; e < kNP2 * kKP2; e += 128) {          // WT1[n][k]
      const int n = e >> 5, k = e & 31;
      WT1[n][k] = (n < kN2 && k < kK2) ? W1t[k * kN2 + n] : 0.0f;
    }
    for (int e = tid; e < kMP * kKP3; e += 128) {           // WT2[n][k]
      const int n = e >> 6, k = e & 63;
      WT2[n][k] = (n < kM && k < kK3) ? W2t[k * kM + n] : 0.0f;
    }
    for (int e = tid; e < 16 * kSP; e += 128) {             // RT[c][s]
      const int c = e / kSP, k = e - c * kSP;
      RT[c][k] = (c < 4 && k < st) ? DM[(sec + k) * 4 + c] : 0.0f;
    }
    for (int e = tid; e < kNP2; e += 128) BB1[e] = (e < kN2) ? B1t[e] : 0.0f;
    for (int e = tid; e < kMP;  e += 128) BB2[e] = (e < kM)  ? B2t[e] : 0.0f;

    // ---- layer 1: X1[s][n] = tanh(ss[s]*W0[n] + b0[n]) (no residual, K=1)
    for (int e = tid; e < spad * kKP2; e += 128) {
      const int s = e >> 5;          // kKP2 == 32
      const int n = e & 31;
      float v = 0.0f;
      if (s < st && n < kK2) {
        const float ss = DM[(sec + s) * 4 + 0];
        v = tanhf(ss * W0t[n] + B0t[n]);
      }
      X1[s][n] = v;
    }
    __syncthreads();

    // ---- layer 2: X2 = tanh(X1 @ W1 + b1) + [X1,X1]   (M=spad, N=64, K=32)
    for (int q = wave; q < mt * 4; q += 4) {
      const int mi = q >> 2;
      const int ni = q & 3;
      const int m  = mi * 16 + l16;
      const int n  = ni * 16 + l16;
      v8f acc = {0.f, 0.f, 0.f, 0.f, 0.f, 0.f, 0.f, 0.f};
      for (int k0 = 0; k0 < kKP2; k0 += 4) {
        const int ka = k0 + 2 * hi;
        v2f a; a.x = X1[m][ka];   a.y = X1[m][ka + 1];
        v2f b; b.x = WT1[n][ka];  b.y = WT1[n][ka + 1];
        acc = wmma_f32(a, b, acc);
      }
      const float bn = BB1[n];
      const int   ri = ((n < kK2) ? n : (n - kK2)) & 31;    // in-bounds always
#pragma unroll
      for (int r = 0; r < 8; ++r) {
        const int mr  = mi * 16 + r + 8 * hi;
        const float res = X1[mr][ri];
        const float v = (mr < st && n < kN2) ? (tanhf(acc[r] + bn) + res) : 0.0f;
        X2[mr][n] = v;
      }
    }
    __syncthreads();

    // ---- layer 3 fused with xyz^T accumulation, one 16-col gg tile at a time
    for (int ni = 0; ni < kMP / 16; ++ni) {
      // gg tile (transposed): GTT[c][s] = gg[s][ni*16+c]   (M=spad, N=16, K=64)
      for (int mi = wave; mi < mt; mi += 4) {
        const int m = mi * 16 + l16;
        const int n = ni * 16 + l16;
        v8f acc = {0.f, 0.f, 0.f, 0.f, 0.f, 0.f, 0.f, 0.f};
        for (int k0 = 0; k0 < kKP3; k0 += 4) {
          const int ka = k0 + 2 * hi;
          v2f a; a.x = X2[m][ka];   a.y = X2[m][ka + 1];
          v2f b; b.x = WT2[n][ka];  b.y = WT2[n][ka + 1];
          acc = wmma_f32(a, b, acc);
        }
        const float bn = BB2[n];
        const int   ri = (n < kK3) ? n : (n - kK3);         // < 64, in-bounds
#pragma unroll
        for (int r = 0; r < 8; ++r) {
          const int mr  = mi * 16 + r + 8 * hi;
          const float res = X2[mr][ri];
          const float v = (mr < st && n < kM) ? (tanhf(acc[r] + bn) + res) : 0.0f;
          GTT[l16][mr] = v;
        }
      }
      __syncthreads();

      // xyz^T tile: XT[ni*16+m][c] += sum_s gg[s][ni*16+m] * rr[s][c]
      // A[m][k] = GTT[m][k], B[k][c] = RT[c][k]  (both contiguous b64 reads)
      if (wave == (ni & 3)) {
        v8f acc = {0.f, 0.f, 0.f, 0.f, 0.f, 0.f, 0.f, 0.f};
        for (int k0 = 0; k0 < spad; k0 += 4) {
          const int ka = k0 + 2 * hi;
          v2f a; a.x = GTT[l16][ka]; a.y = GTT[l16][ka + 1];
          v2f b; b.x = RT[l16][ka];  b.y = RT[l16][ka + 1];
          acc = wmma_f32(a, b, acc);
        }
#pragma unroll
        for (int r = 0; r < 8; ++r) {
          const int mr = ni * 16 + r + 8 * hi;
          XT[mr][l16] += acc[r];
        }
      }
      __syncthreads();
    }
  }

  // ---------------------------------------------------- scale xyz by 1/NNEI
  for (int e = tid; e < kMP * 16; e += 128)
    (&XT[0][0])[e] *= (1.0f / (float)kNNEI);
  __syncthreads();

  // ---- res[m][a] = sum_c XT[m][c]*XT[a][c]   (M=112, N=16, K=4: one WMMA/tile)
  for (int mi = wave; mi < kMP / 16; mi += 4) {
    const int ka = 2 * hi;
    v2f a; a.x = XT[mi * 16 + l16][ka]; a.y = XT[mi * 16 + l16][ka + 1];
    v2f b; b.x = XT[l16][ka];           b.y = XT[l16][ka + 1];
    v8f acc = {0.f, 0.f, 0.f, 0.f, 0.f, 0.f, 0.f, 0.f};
    acc = wmma_f32(a, b, acc);
#pragma unroll
    for (int r = 0; r < 8; ++r) {
      const int mr = mi * 16 + r + 8 * hi;
      if (mr < kM)
        out[(size_t)g * (kM * kAXIS) + (size_t)mr * kAXIS + l16] = acc[r];
    }
  }
}

// ---------------------------------------------------------------------------
extern "C" void kernel_launch(void* const* d_in, const int* in_sizes, int n_in,
                              void* d_out, int out_size, void* d_ws, size_t ws_size,
                              hipStream_t stream) {
  (void)in_sizes; (void)n_in; (void)out_size; (void)d_ws; (void)ws_size;
  const int*   nlist  = (const int*)  d_in[0];
  const float* coord  = (const float*)d_in[1];
  const int*   atype  = (const int*)  d_in[2];
  const float* mean   = (const float*)d_in[3];
  const float* stdv   = (const float*)d_in[4];
  const float* w0     = (const float*)d_in[5];
  const float* b0     = (const float*)d_in[6];
  const float* w1     = (const float*)d_in[7];
  const float* b1     = (const float*)d_in[8];
  const float* w2     = (const float*)d_in[9];
  const float* b2     = (const float*)d_in[10];
  float* out = (float*)d_out;

  dim3 grid(kNF * kNLOC);   // one workgroup per local atom
  dim3 block(128);          // 4 wave32
  descrpt_sea_fused<<<grid, block, 0, stream>>>(nlist, coord, atype, mean, stdv,
                                                w0, b0, w1, b1, w2, b2, out);
}